// PatchApplier_63685775065303
// MI455X (gfx1250) — compile-verified
//
#include <hip/hip_runtime.h>
#include <hip/hip_bf16.h>

// Problem constants (match reference)
#define IMG_H 2048
#define IMG_W 2048
#define IMG_PLANE (IMG_H * IMG_W)
#define N_ANN 64
#define PATCH 512
#define PATCH_PLANE (PATCH * PATCH)
#define TILE_W 64
#define TILE_H 16
#define KMAX 144            // max padded K rows for the WMMA path (wh>=64 -> Kr<=138)

typedef float v2f_ __attribute__((ext_vector_type(2)));
typedef float v8f_ __attribute__((ext_vector_type(8)));

// Types matching the async builtin signature reported by clang:
//   param0: int __attribute__((vector_size(16))) __device__ *  (addrspace(1), non-const)
typedef int v4i_ __attribute__((vector_size(16)));
typedef __attribute__((address_space(1))) v4i_* g_v4i;
typedef __attribute__((address_space(3))) v4i_* l_v4i;

__device__ __forceinline__ float tri(float s, float j, float scale) {
    return fmaxf(0.0f, 1.0f - fabsf(s - j) * scale);
}

// Exact jax.image.resize(method="bilinear", antialias=True) sample for one
// output pixel (ix,iy) of a wh x wh resize of the 512x512 patch, all 3 channels.
__device__ __forceinline__ void sample3(const float* __restrict__ patch,
                                        int ix, int iy, int wh,
                                        float& r0, float& r1, float& r2) {
    const float inv   = 512.0f / (float)wh;   // kernel_scale = 1/scale (>1: downscale)
    const float scale = (float)wh / 512.0f;
    const float sx = ((float)ix + 0.5f) * inv - 0.5f;
    const float sy = ((float)iy + 0.5f) * inv - 0.5f;

    int jx0 = (int)ceilf(sx - inv);  if (jx0 < 0)   jx0 = 0;
    int jx1 = (int)floorf(sx + inv); if (jx1 > 511) jx1 = 511;
    int jy0 = (int)ceilf(sy - inv);  if (jy0 < 0)   jy0 = 0;
    int jy1 = (int)floorf(sy + inv); if (jy1 > 511) jy1 = 511;

    float swx = 0.0f, swy = 0.0f;
    for (int j = jx0; j <= jx1; ++j) swx += tri(sx, (float)j, scale);
    for (int j = jy0; j <= jy1; ++j) swy += tri(sy, (float)j, scale);

    float a0 = 0.0f, a1 = 0.0f, a2 = 0.0f;
    for (int jy = jy0; jy <= jy1; ++jy) {
        const float wy = tri(sy, (float)jy, scale);
        const float* __restrict__ row = patch + (size_t)jy * PATCH;
        float t0 = 0.0f, t1 = 0.0f, t2 = 0.0f;
        for (int jx = jx0; jx <= jx1; ++jx) {
            const float wx = tri(sx, (float)jx, scale);
            t0 += wx * row[jx];
            t1 += wx * row[jx + PATCH_PLANE];
            t2 += wx * row[jx + 2 * PATCH_PLANE];
        }
        a0 += wy * t0; a1 += wy * t1; a2 += wy * t2;
    }
    const float norm = 1.0f / (swx * swy);
    r0 = a0 * norm; r1 = a1 * norm; r2 = a2 * norm;
}

__global__ __launch_bounds__(256) void patch_apply_kernel(
    const float* __restrict__ img,
    const float* __restrict__ ann,
    const float* __restrict__ patch,
    float* __restrict__ out) {

    __shared__ int s_ax[N_ANN];
    __shared__ int s_ay[N_ANN];
    __shared__ int s_wh[N_ANN];
    __shared__ int s_list[N_ANN];
    __shared__ int s_n;
    // 12 KB buffer: async staging (copy path) OR M matrix (WMMA path) — disjoint paths
    __shared__ alignas(16) float s_shared[3 * TILE_W * TILE_H];
    __shared__ float s_swx[16];
    __shared__ float s_swy[16];

    const int tid    = threadIdx.x;
    const int tile_x = blockIdx.x * TILE_W;
    const int tile_y = blockIdx.y * TILE_H;

    // --- per-block geometry (exact reference integer semantics) ---
    if (tid < N_ANN) {
        const float x1f = ann[tid * 4 + 0] * (float)IMG_W;
        const float y1f = ann[tid * 4 + 1] * (float)IMG_H;
        const float x2f = ann[tid * 4 + 2] * (float)IMG_W;
        const float y2f = ann[tid * 4 + 3] * (float)IMG_H;
        const int x1 = (int)x1f, y1 = (int)y1f, x2 = (int)x2f, y2 = (int)y2f;
        const int bw = x2 - x1, bh = y2 - y1;
        const int mn = bw < bh ? bw : bh;
        int wh = mn / 2;            // int(min*0.5) == floor for non-negative
        if (wh < 10) wh = 10;
        int x = x1 + bw / 2 - wh / 2;
        int y = y1 + bh / 2 - wh / 2;
        if (y > IMG_H - wh) y = IMG_H - wh;  if (y < 0) y = 0;
        if (x > IMG_W - wh) x = IMG_W - wh;  if (x < 0) x = 0;
        s_ax[tid] = x; s_ay[tid] = y; s_wh[tid] = wh;
    }
    __syncthreads();

    // --- cull: list of boxes intersecting this tile (ascending index) ---
    if (tid == 0) {
        int n = 0;
        for (int a = 0; a < N_ANN; ++a) {
            if (s_ax[a] < tile_x + TILE_W && s_ax[a] + s_wh[a] > tile_x &&
                s_ay[a] < tile_y + TILE_H && s_ay[a] + s_wh[a] > tile_y) {
                s_list[n++] = a;
            }
        }
        s_n = n;
    }
    __syncthreads();
    const int n = s_n;  // block-uniform

    // --- WMMA fast-path eligibility (all block-uniform arithmetic) ---
    // Topmost-possible box == highest intersecting index; if it covers the whole
    // tile, every pixel's topmost box is it.
    bool fast = false;
    int ax = 0, ay = 0, wh = 0;
    int Jy0 = 0, Kr = 0, KrPad = 0;
    float inv = 0.0f, scale = 0.0f;
    if (n > 0) {
        const int A = s_list[n - 1];
        ax = s_ax[A]; ay = s_ay[A]; wh = s_wh[A];
        if (wh >= 64 &&
            ax <= tile_x && ax + wh >= tile_x + TILE_W &&
            ay <= tile_y && ay + wh >= tile_y + TILE_H) {
            inv   = 512.0f / (float)wh;
            scale = (float)wh / 512.0f;
            const float sy0  = ((float)(tile_y - ay) + 0.5f) * inv - 0.5f;
            const float sy15 = ((float)(tile_y + 15 - ay) + 0.5f) * inv - 0.5f;
            int j0 = (int)ceilf(sy0 - inv);   if (j0 < 0)   j0 = 0;
            int j1 = (int)floorf(sy15 + inv); if (j1 > 511) j1 = 511;
            Jy0 = j0;
            Kr  = j1 - j0 + 1;
            KrPad = (Kr + 3) & ~3;
            fast = (KrPad <= KMAX);
        }
    }

    const int lx = (tid & 15) * 4;   // 0..60, float4 granularity
    const int ly = tid >> 4;         // 0..15
    const size_t base = (size_t)(tile_y + ly) * IMG_W + (tile_x + lx);

    if (n == 0) {
        // ---- pure streaming copy: CDNA5 async LDS-bypass path ----
        for (int c = 0; c < 3; ++c) {
            float* g = const_cast<float*>(img) + base + (size_t)c * IMG_PLANE;
            __builtin_amdgcn_global_load_async_to_lds_b128(
                (g_v4i)g, (l_v4i)&s_shared[c * 1024 + tid * 4], 0, 0);
        }
#if __has_builtin(__builtin_amdgcn_s_wait_asynccnt)
        __builtin_amdgcn_s_wait_asynccnt(0);
#else
        asm volatile("s_wait_asynccnt 0" ::: "memory");
#endif
        for (int c = 0; c < 3; ++c) {
            float* g = out + base + (size_t)c * IMG_PLANE;
            __builtin_amdgcn_global_store_async_from_lds_b128(
                (g_v4i)g, (l_v4i)&s_shared[c * 1024 + tid * 4], 0, 0);
        }
        // S_ENDPGM performs an implicit wait-idle; ASYNCcnt drains before exit.
    } else if (fast) {
        // ---- whole tile = one box: resize as f32 WMMA matmul ----
        // D(16x16) = Wy(16xKr) x M(Krx16), per 16x16 sub-tile and channel.
        if (tid < 16) {   // swy shared by all 4 sub-tiles (same output rows)
            const float sy = ((float)(tile_y + tid - ay) + 0.5f) * inv - 0.5f;
            int j0 = (int)ceilf(sy - inv);  if (j0 < 0)   j0 = 0;
            int j1 = (int)floorf(sy + inv); if (j1 > 511) j1 = 511;
            float s = 0.0f;
            for (int j = j0; j <= j1; ++j) s += tri(sy, (float)j, scale);
            s_swy[tid] = s;
        }
        for (int c = 0; c < 3; ++c) {
            const float* __restrict__ pc = patch + (size_t)c * PATCH_PLANE;
            for (int s = 0; s < 4; ++s) {
                const int tx16 = tile_x + s * 16;
                __syncthreads();   // M/swx safe to (re)write
                if (tid < 16) {
                    const float sx = ((float)(tx16 + tid - ax) + 0.5f) * inv - 0.5f;
                    int j0 = (int)ceilf(sx - inv);  if (j0 < 0)   j0 = 0;
                    int j1 = (int)floorf(sx + inv); if (j1 > 511) j1 = 511;
                    float sw = 0.0f;
                    for (int j = j0; j <= j1; ++j) sw += tri(sx, (float)j, scale);
                    s_swx[tid] = sw;
                }
                // M[jy][u] = sum_jx Wx[u,jx] * P[c, Jy0+jy, jx]; zero-pad jy>=Kr
                for (int e = tid; e < KrPad * 16; e += 256) {
                    const int jy = e >> 4, u = e & 15;
                    float m = 0.0f;
                    if (jy < Kr) {
                        const float sx = ((float)(tx16 + u - ax) + 0.5f) * inv - 0.5f;
                        int j0 = (int)ceilf(sx - inv);  if (j0 < 0)   j0 = 0;
                        int j1 = (int)floorf(sx + inv); if (j1 > 511) j1 = 511;
                        const float* __restrict__ prow = pc + (size_t)(Jy0 + jy) * PATCH;
                        for (int j = j0; j <= j1; ++j) m += tri(sx, (float)j, scale) * prow[j];
                    }
                    s_shared[e] = m;
                }
                __syncthreads();
                if (tid < 32) {    // one full wave: EXEC all-ones for WMMA
                    const int  col = tid & 15;
                    const bool hiH = tid >= 16;
                    // A rows: lanes 0-15 and 16-31 both index rows 0-15
                    const float sym = ((float)(tile_y + col - ay) + 0.5f) * inv - 0.5f;
                    v8f_ acc = {};
                    for (int kb = 0; kb < KrPad; kb += 4) {
                        const int kxi = kb + (hiH ? 2 : 0);
                        const int kyi = kb + (hiH ? 3 : 1);
                        v2f_ a, b;
                        a.x = tri(sym, (float)(Jy0 + kxi), scale);  // ==0 past Kr
                        a.y = tri(sym, (float)(Jy0 + kyi), scale);
                        b.x = s_shared[kxi * 16 + col];
                        b.y = s_shared[kyi * 16 + col];
                        acc = __builtin_amdgcn_wmma_f32_16x16x4_f32(
                                  false, a, false, b, (short)0, acc, false, false);
                    }
                    const float rswx = 1.0f / s_swx[col];
                    const int   rb   = hiH ? 8 : 0;   // D rows: VGPR r -> row r + 8*hi
                    #pragma unroll
                    for (int r = 0; r < 8; ++r) {
                        const int row = rb + r;
                        const float v = acc[r] * rswx / s_swy[row];
                        out[(size_t)(tile_y + row) * IMG_W + (tx16 + col)
                            + (size_t)c * IMG_PLANE] = v;
                    }
                }
            }
        }
    } else {
        // ---- mixed tile: per-pixel topmost-box resolve (exact scalar path) ----
        const int gx = tile_x + lx;
        const int gy = tile_y + ly;
        for (int q = 0; q < 4; ++q) {
            const int px = gx + q;
            const int py = gy;
            int hit = -1;
            for (int k = n - 1; k >= 0; --k) {   // highest index wins (last paste)
                const int a = s_list[k];
                if (px >= s_ax[a] && px < s_ax[a] + s_wh[a] &&
                    py >= s_ay[a] && py < s_ay[a] + s_wh[a]) { hit = a; break; }
            }
            const size_t idx = (size_t)py * IMG_W + px;
            if (hit < 0) {
                out[idx]                 = img[idx];
                out[idx + IMG_PLANE]     = img[idx + IMG_PLANE];
                out[idx + 2 * IMG_PLANE] = img[idx + 2 * IMG_PLANE];
            } else {
                float r0, r1, r2;
                sample3(patch, px - s_ax[hit], py - s_ay[hit], s_wh[hit], r0, r1, r2);
                out[idx]                 = r0;
                out[idx + IMG_PLANE]     = r1;
                out[idx + 2 * IMG_PLANE] = r2;
            }
        }
    }
}

extern "C" void kernel_launch(void* const* d_in, const int* in_sizes, int n_in,
                              void* d_out, int out_size, void* d_ws, size_t ws_size,
                              hipStream_t stream) {
    const float* img   = (const float*)d_in[0];   // [3,2048,2048]
    const float* ann   = (const float*)d_in[1];   // [64,4]
    const float* patch = (const float*)d_in[2];   // [3,512,512]
    float* out = (float*)d_out;                   // [3,2048,2048]

    dim3 grid(IMG_W / TILE_W, IMG_H / TILE_H);    // 32 x 128 tiles
    patch_apply_kernel<<<grid, 256, 0, stream>>>(img, ann, patch, out);
}